// ContrastMSELoss_37563783971102
// MI455X (gfx1250) — compile-verified
//
#include <hip/hip_runtime.h>
#include <math.h>

// ---------------------------------------------------------------------------
// ContrastMSELoss for MI455X (gfx1250, wave32, WMMA + async global->LDS)
//   out = MSE(seg_thick,target_thick) + MSE(seg_thin,target_thin)
//       + 0.1 * (contrastive_thick + contrastive_thin)
// Contrastive GEMM uses V_WMMA_F32_16X16X4_F32 (native f32 matrix path).
// B tiles are double-buffered with GLOBAL_LOAD_ASYNC_TO_LDS_B128 (ASYNCcnt),
// so the single wave per block overlaps L2->LDS copies with WMMA compute.
// Zero half of the negative cache is folded in analytically (raw logit == 0).
// ---------------------------------------------------------------------------

typedef __attribute__((ext_vector_type(2))) float v2f;
typedef __attribute__((ext_vector_type(8))) float v8f;

#define TEMP      0.1f
#define BASET     0.07f
#define LOSSW     0.1f
#define INV_T     10.0f           // 1/TEMP
#define A_CNT     16
#define V_CNT     64
#define D_CNT     256
#define CACHE_N   10000
#define NROWS     1024            // A*V
#define NTJ       (CACHE_N / 16)  // 625 column tiles
#define LDST      260             // padded LDS row stride (floats) vs bank conflicts
#define MSE_N     2097152         // 8*1*512*512 elements per map

// 16-lane (half-wave) reductions; rows of the WMMA C tile live on 16 lanes.
__device__ inline float redmax16(float x) {
  #pragma unroll
  for (int m = 8; m >= 1; m >>= 1) x = fmaxf(x, __shfl_xor(x, m, 32));
  return x;
}
__device__ inline float redsum16(float x) {
  #pragma unroll
  for (int m = 8; m >= 1; m >>= 1) x += __shfl_xor(x, m, 32);
  return x;
}

// Async-copy one B-tile slice (this lane's 128 floats = 32 x B128) into LDS.
// vdst operand = per-lane LDS byte address; vaddr = per-lane 64-bit global addr.
__device__ inline void async_stage_row(const float* __restrict__ src,
                                       unsigned lds_byte_off) {
  #pragma unroll
  for (int d = 0; d < 128; d += 4) {
    asm volatile("global_load_async_to_lds_b128 %0, %1, off"
                 :: "v"(lds_byte_off + (unsigned)(d * 4)),
                    "v"(src + d)
                 : "memory");
  }
}
// Wait until the OLDER tile's 32 async ops are done (loads retire in order).
__device__ inline void wait_async_older() {
  asm volatile("s_wait_asynccnt 0x20" ::: "memory");
}
__device__ inline void wait_async_all() {
  asm volatile("s_wait_asynccnt 0x0" ::: "memory");
}

// ---------------------------------------------------------------------------
// Contrastive kernel: grid = (64 row-tiles, 2 sides), block = 1 wave (32).
// Row tile t covers anchor rows r = 16t..16t+15  ->  view v = t, anchor a = i.
// ---------------------------------------------------------------------------
__global__ __launch_bounds__(32)
void contrastive_kernel(const float* __restrict__ Xthick,
                        const float* __restrict__ Xthin,
                        const float* __restrict__ Qthick,
                        const float* __restrict__ Qthin,
                        const int*   __restrict__ ythick,
                        const int*   __restrict__ ythin,
                        float* __restrict__ accum) {
  const int t = blockIdx.x;                         // row tile == view index
  const float* X  = blockIdx.y ? Xthin  : Xthick;
  const float* Q1 = (blockIdx.y ? Qthin : Qthick) + (size_t)CACHE_N * D_CNT; // class-1 rows
  const int*   y  = blockIdx.y ? ythin  : ythick;

  __shared__ float ldsA[16 * LDST];
  __shared__ float ldsB[2][16 * LDST];              // double buffer for async staging

  const int lane = threadIdx.x;        // 0..31
  const int mrow = lane & 15;          // row (A) / column (B) within tile
  const int hi   = lane >> 4;          // K-pair selector per WMMA f32 layout

  // Per-lane LDS byte offsets of the two B buffers (low 32 bits of flat shared
  // pointer == LDS offset on AMDGPU).
  const unsigned bOff0 =
      (unsigned)(uintptr_t)&ldsB[0][mrow * LDST + hi * 128];
  const unsigned bOff1 =
      (unsigned)(uintptr_t)&ldsB[1][mrow * LDST + hi * 128];

  // ---- stage A tile: ldsA[i][d] = X[a=i][v=t][d] --------------------------
  {
    const float* src = X + ((size_t)mrow * V_CNT + t) * D_CNT + hi * 128;
    float*       dst = ldsA + mrow * LDST + hi * 128;
    #pragma unroll 4
    for (int d = 0; d < 128; d += 4)
      *(float4*)(dst + d) = *(const float4*)(src + d);
  }
  __syncthreads();

  const float* pA  = ldsA + mrow * LDST + 2 * hi;
  const float* pB0 = &ldsB[0][mrow * LDST + 2 * hi];
  const float* pB1 = &ldsB[1][mrow * LDST + 2 * hi];
  const float* qrow = Q1 + (size_t)mrow * D_CNT + hi * 128;   // + tj*16*D per tile

  // ---- PASS 1: online row max m[] and exp-sum E[] over the 10000 real cols.
  // m starts at 0.0 == raw logit of every zero column (covers their max).
  float m[8], E[8];
  #pragma unroll
  for (int v = 0; v < 8; ++v) { m[v] = 0.0f; E[v] = 0.0f; }

  async_stage_row(qrow, bOff0);                     // prefetch tile 0 -> buf 0
  for (int tj = 0; tj < NTJ; ++tj) {
    const int buf = tj & 1;
    if (tj + 1 < NTJ)                               // prefetch next tile
      async_stage_row(qrow + (size_t)(tj + 1) * 16 * D_CNT, (buf ^ 1) ? bOff1 : bOff0);
    if (tj + 1 < NTJ) wait_async_older(); else wait_async_all();

    const float* pB = buf ? pB1 : pB0;
    v8f acc = {};
    #pragma unroll 8
    for (int kk = 0; kk < D_CNT; kk += 4) {
      v2f a = *(const v2f*)(pA + kk);
      v2f b = *(const v2f*)(pB + kk);
      acc = __builtin_amdgcn_wmma_f32_16x16x4_f32(
                false, a, false, b, (short)0, acc, false, false);
    }

    #pragma unroll
    for (int v = 0; v < 8; ++v) {
      float lv = acc[v] * INV_T;                    // logits / TEMPERATURE
      float tm = redmax16(lv);
      float mn = fmaxf(m[v], tm);
      float s  = redsum16(__expf(lv - mn));
      E[v] = E[v] * __expf(m[v] - mn) + s;
      m[v] = mn;
    }
  }

  // ---- per-row constants ---------------------------------------------------
  // y==1 rows: negatives are the 10000 zero cols -> S_neg = 10000*exp(-m),
  //            positives are real cols minus the diagonal (count 9999).
  // y==0 rows: negatives are the real cols (S_neg = E), positives are the
  //            10000 identical zero cols -> mean term computed analytically.
  float Sneg[8], czero[8];
  int   ypos[8];
  #pragma unroll
  for (int v = 0; v < 8; ++v) {
    int i   = v + 8 * hi;                 // anchor index a for this row
    ypos[v] = y[i];
    float em = __expf(-m[v]);
    Sneg[v]  = (float)CACHE_N * em;
    czero[v] = (-m[v]) - __logf(em + E[v]);
  }

  // ---- PASS 2: recompute logits, accumulate positive log-prob sums --------
  float pos[8];
  #pragma unroll
  for (int v = 0; v < 8; ++v) pos[v] = 0.0f;

  async_stage_row(qrow, bOff0);                     // prefetch tile 0 -> buf 0
  for (int tj = 0; tj < NTJ; ++tj) {
    const int buf = tj & 1;
    if (tj + 1 < NTJ)
      async_stage_row(qrow + (size_t)(tj + 1) * 16 * D_CNT, (buf ^ 1) ? bOff1 : bOff0);
    if (tj + 1 < NTJ) wait_async_older(); else wait_async_all();

    const float* pB = buf ? pB1 : pB0;
    v8f acc = {};
    #pragma unroll 8
    for (int kk = 0; kk < D_CNT; kk += 4) {
      v2f a = *(const v2f*)(pA + kk);
      v2f b = *(const v2f*)(pB + kk);
      acc = __builtin_amdgcn_wmma_f32_16x16x4_f32(
                false, a, false, b, (short)0, acc, false, false);
    }

    #pragma unroll
    for (int v = 0; v < 8; ++v) {
      float l    = acc[v] * INV_T - m[v];
      float term = l - __logf(__expf(l) + Sneg[v]);
      // diagonal exclusion: global col j = 16*tj + mrow equals row r = 16*t + i
      bool  diag = (tj == t) && (mrow == (v + 8 * hi));
      bool  take = (ypos[v] == 1) && !diag;
      pos[v] += take ? term : 0.0f;
    }
  }

  // ---- finalize: per-row loss, sum over the tile's 16 rows ----------------
  const float c = -(TEMP / BASET);
  float s = 0.0f;
  #pragma unroll
  for (int v = 0; v < 8; ++v) {
    float ps = redsum16(pos[v]);                          // row positive sum
    float lr = (ypos[v] == 1) ? (ps * (1.0f / 9999.0f))   // mean over 9999 pos
                              : czero[v];                 // identical zero-col term
    s += c * lr;
  }
  float tot = s + __shfl_xor(s, 16, 32);   // rows 0-7 half + rows 8-15 half
  if (lane == 0) atomicAdd(accum, tot);
}

// ---------------------------------------------------------------------------
// MSE reduction over both (seg,target) pairs, float4 grid-stride.
// ---------------------------------------------------------------------------
__global__ void mse_kernel(const float* __restrict__ a0, const float* __restrict__ b0,
                           const float* __restrict__ a1, const float* __restrict__ b1,
                           float* __restrict__ accum) {
  const int n4 = MSE_N / 4;                 // 524288 float4 per pair
  float s = 0.0f;
  int stride = gridDim.x * blockDim.x;
  for (int i = blockIdx.x * blockDim.x + threadIdx.x; i < 2 * n4; i += stride) {
    const float4* pa = (i < n4) ? (const float4*)a0 : (const float4*)a1;
    const float4* pb = (i < n4) ? (const float4*)b0 : (const float4*)b1;
    int j = (i < n4) ? i : i - n4;
    float4 x = pa[j], yv = pb[j];
    float dx = x.x - yv.x, dy = x.y - yv.y, dz = x.z - yv.z, dw = x.w - yv.w;
    s += dx * dx + dy * dy + dz * dz + dw * dw;
  }
  #pragma unroll
  for (int mask = 16; mask >= 1; mask >>= 1) s += __shfl_xor(s, mask, 32);
  if ((threadIdx.x & 31) == 0) atomicAdd(accum, s);
}

__global__ void init_kernel(float* ws) { ws[0] = 0.0f; ws[1] = 0.0f; }

__global__ void final_kernel(const float* __restrict__ ws, float* __restrict__ out) {
  out[0] = ws[0] * (1.0f / (float)MSE_N) + LOSSW * ws[1] * (1.0f / (float)NROWS);
}

// ---------------------------------------------------------------------------
extern "C" void kernel_launch(void* const* d_in, const int* in_sizes, int n_in,
                              void* d_out, int out_size, void* d_ws, size_t ws_size,
                              hipStream_t stream) {
  const float* seg_thick    = (const float*)d_in[0];
  const float* seg_thin     = (const float*)d_in[1];
  const float* target_thick = (const float*)d_in[2];
  const float* target_thin  = (const float*)d_in[3];
  const float* X_thick      = (const float*)d_in[4];
  const float* X_thin       = (const float*)d_in[5];
  const float* Q_thick      = (const float*)d_in[6];
  const float* Q_thin       = (const float*)d_in[7];
  const int*   y_thick      = (const int*)d_in[8];
  const int*   y_thin       = (const int*)d_in[9];
  float* ws  = (float*)d_ws;
  float* out = (float*)d_out;

  init_kernel<<<1, 1, 0, stream>>>(ws);

  mse_kernel<<<1024, 256, 0, stream>>>(seg_thick, target_thick,
                                       seg_thin,  target_thin, &ws[0]);

  dim3 cgrid(NROWS / 16, 2);   // 64 row tiles x {thick, thin}
  contrastive_kernel<<<cgrid, 32, 0, stream>>>(X_thick, X_thin,
                                               Q_thick, Q_thin,
                                               y_thick, y_thin, &ws[1]);

  final_kernel<<<1, 1, 0, stream>>>(ws, out);
}